// GeoAttention_40286793236552
// MI455X (gfx1250) — compile-verified
//
#include <hip/hip_runtime.h>
#include <math.h>

// B=2048, N=128, R=512
#define BATCH 2048
#define NN    128
#define RR    512

typedef __attribute__((ext_vector_type(2))) float v2f;
typedef __attribute__((ext_vector_type(4))) float v4f;
typedef __attribute__((ext_vector_type(8))) float v8f;

#define LDA 130  // padded LDS row stride (floats) to spread banks

// ---------------------------------------------------------------------------
// Phase 1: weight = softmax( exp(-50*d) @ K + bias ), one block = 16 batch rows
// Uses V_WMMA_F32_16X16X4_F32 (full fp32 matrix core path on gfx1250).
// ---------------------------------------------------------------------------
__global__ __launch_bounds__(256) void geo_phase1(
    const float* __restrict__ dist,     // [BATCH, NN]
    const float* __restrict__ kmat,     // [NN, NN]
    const float* __restrict__ bias,     // [NN]
    float* __restrict__ weight)         // [BATCH, NN]  (workspace)
{
    __shared__ float sA[16 * LDA];   // simi tile (later reused for exp values)
    __shared__ float sL[16 * LDA];   // logits tile
    __shared__ float sInv[16];

    const int b0  = blockIdx.x * 16;
    const int tid = threadIdx.x;

    // simi = exp(-50*d) for 16x128 tile
    for (int i = tid; i < 16 * NN; i += 256) {
        const int m = i >> 7, k = i & 127;
        sA[m * LDA + k] = expf(-50.0f * dist[(size_t)(b0 + m) * NN + k]);
    }
    __syncthreads();

    const int wave = tid >> 5;          // 0..7 -> output N-tile
    const int lane = tid & 31;
    const int hf   = lane >> 4;         // lane half (0/1)
    const int lm   = lane & 15;         // A row / B,C column within tile
    const int col  = wave * 16 + lm;    // global output column (0..127)

    v8f acc = {};
    for (int k0 = 0; k0 < NN; k0 += 4) {
        // A fragment 16x4: VGPR v, half h -> K = k0 + v + 2h, M = lm
        v2f a, b;
        a.x = sA[lm * LDA + k0 + 2 * hf + 0];
        a.y = sA[lm * LDA + k0 + 2 * hf + 1];
        // B fragment 4x16: VGPR v, half h -> row K = k0 + v + 2h, col = lm
        b.x = kmat[(size_t)(k0 + 2 * hf + 0) * NN + col];
        b.y = kmat[(size_t)(k0 + 2 * hf + 1) * NN + col];
        acc = __builtin_amdgcn_wmma_f32_16x16x4_f32(
            false, a, false, b, (short)0, acc, false, false);
    }

    // D layout: VGPR v -> row m = v + 8*half, col = lm (within tile)
    const float bc = bias[col];
    #pragma unroll
    for (int v = 0; v < 8; ++v) {
        const int m = v + 8 * hf;
        sL[m * LDA + col] = acc[v] + bc;
    }
    __syncthreads();   // all waves done with sA + logits visible

    // Row softmax: one lane per row (cost negligible vs phase 2 stream)
    if (tid < 16) {
        const int m = tid;
        float mx = -INFINITY;
        for (int n = 0; n < NN; ++n) mx = fmaxf(mx, sL[m * LDA + n]);
        float sum = 0.0f;
        for (int n = 0; n < NN; ++n) {
            const float e = expf(sL[m * LDA + n] - mx);
            sA[m * LDA + n] = e;        // sA reused as exp scratch
            sum += e;
        }
        sInv[m] = 1.0f / sum;
    }
    __syncthreads();

    // Coalesced write of normalized weights
    for (int i = tid; i < 16 * NN; i += 256) {
        const int m = i >> 7, n = i & 127;
        weight[(size_t)(b0 + m) * NN + n] = sA[m * LDA + n] * sInv[m];
    }
}

// ---------------------------------------------------------------------------
// Phase 2: out[b,r] = sum_n weight[b,n] * ctx[b,n,r]
// One block per batch element; 128 threads x float4 = 512 r-columns.
// Streams 512 MiB of context once -> HBM-bandwidth bound (~23 us floor).
// ---------------------------------------------------------------------------
__global__ __launch_bounds__(128) void geo_phase2(
    const float* __restrict__ weight,   // [BATCH, NN]
    const float* __restrict__ ctx,      // [BATCH, NN, RR]
    float* __restrict__ out)            // [BATCH, RR]
{
    __shared__ float w[NN];
    const int b   = blockIdx.x;
    const int tid = threadIdx.x;

    w[tid] = weight[(size_t)b * NN + tid];
    __syncthreads();

    const v4f* cp = (const v4f*)(ctx + (size_t)b * NN * RR) + tid; // [n][RR/4]
    v4f acc = {0.0f, 0.0f, 0.0f, 0.0f};

    #pragma unroll 8
    for (int n = 0; n < NN; ++n) {
        const v4f c = __builtin_nontemporal_load(&cp[(size_t)n * (RR / 4)]);
        const float wn = w[n];
        acc.x += wn * c.x;
        acc.y += wn * c.y;
        acc.z += wn * c.z;
        acc.w += wn * c.w;
    }

    v4f* op = (v4f*)(out + (size_t)b * RR) + tid;
    *op = acc;
}

// ---------------------------------------------------------------------------
extern "C" void kernel_launch(void* const* d_in, const int* in_sizes, int n_in,
                              void* d_out, int out_size, void* d_ws, size_t ws_size,
                              hipStream_t stream) {
    const float* dist = (const float*)d_in[0];   // [2048,128]
    const float* ctx  = (const float*)d_in[1];   // [2048,128,512]
    const float* kmat = (const float*)d_in[2];   // [128,128]
    const float* bias = (const float*)d_in[3];   // [128]
    float* out = (float*)d_out;                  // [2048,512]
    float* wgt = (float*)d_ws;                   // [2048,128] = 1 MiB scratch

    geo_phase1<<<BATCH / 16, 256, 0, stream>>>(dist, kmat, bias, wgt);
    geo_phase2<<<BATCH, 128, 0, stream>>>(wgt, ctx, out);
}